// RelativePositionalEmbedding_86990267613352
// MI455X (gfx1250) — compile-verified
//
#include <hip/hip_runtime.h>
#include <hip/hip_bf16.h>

typedef __attribute__((ext_vector_type(2))) float v2f;
typedef __attribute__((ext_vector_type(4))) float v4f;
typedef __attribute__((ext_vector_type(8))) float v8f;

// Problem constants (from reference setup_inputs):
//   H=16 heads, P=2047 positions, D=64 head_dim, L=2048 length.
#define RPE_H 16
#define RPE_P 2047
#define RPE_D 64
#define RPE_L 2048

// ---------------------------------------------------------------------------
// Phase 1: pos_sums[h][p] = sum_d E[h][p][d], via V_WMMA_F32_16X16X4_F32.
// One wave (32 threads) per 16-position tile. C += A(16x4) * ones(4x16),
// accumulated over 16 K-chunks => every column of C holds the 64-wide row sums.
// B == all-ones makes the B VGPR layout irrelevant.
// ---------------------------------------------------------------------------
__global__ __launch_bounds__(32)
void rpe_reduce_wmma(const float* __restrict__ E, float* __restrict__ pos_sums) {
    const int tile = blockIdx.x;            // H * 128 tiles (ceil(2047/16)=128)
    const int h    = tile >> 7;
    const int p0   = (tile & 127) << 4;
    const int lane = threadIdx.x;           // 0..31 (wave32)
    const int m    = lane & 15;             // row within tile
    const int koff = (lane >> 4) << 1;      // 0 (lanes 0-15) or 2 (lanes 16-31)

    int p  = p0 + m;
    int pc = p < RPE_P ? p : RPE_P - 1;     // clamp so loads stay in-bounds
    const float* base = E + ((size_t)h * RPE_P + pc) * RPE_D + koff;

    v8f c = {0.f, 0.f, 0.f, 0.f, 0.f, 0.f, 0.f, 0.f};
    v2f b = {1.0f, 1.0f};                   // ones matrix (layout-independent)

    #pragma unroll
    for (int k0 = 0; k0 < RPE_D; k0 += 4) {
        // A layout (32-bit 16x4): lane<16 -> {K=k0,K=k0+1}; lane>=16 -> {K=k0+2,K=k0+3}
        v2f a = *(const v2f*)(base + k0);
        c = __builtin_amdgcn_wmma_f32_16x16x4_f32(
                /*neg_a=*/false, a, /*neg_b=*/false, b,
                /*c_mod=*/(short)0, c, /*reuse_a=*/false, /*reuse_b=*/false);
    }

    // C/D layout: VGPR r, lanes 0-15 -> D[M=r][N=lane]; lanes 16-31 -> D[M=8+r].
    // Lane 0 holds sums for rows 0..7 in c[0..7]; lane 16 holds rows 8..15.
    if (lane == 0 || lane == 16) {
        const int mbase = (lane >> 4) << 3;
        #pragma unroll
        for (int r = 0; r < 8; ++r) {
            int pp = p0 + mbase + r;
            if (pp < RPE_P) pos_sums[(size_t)h * RPE_P + pp] = c[r];
        }
    }
}

// ---------------------------------------------------------------------------
// Phase 2: out[h][i][j] = pos_sums[h][clamp(j - i + 1023, 0, 2046)].
// Store-bandwidth bound (256 MB). Each block: one head, 16 rows.
// pos_sums[h] (8 KB) staged in LDS once per block; each thread writes
// 2 float4 non-temporal stores per row (fully coalesced, 16B aligned).
// ---------------------------------------------------------------------------
#define RPE_ROWS 16

__global__ __launch_bounds__(256)
void rpe_gather_rows(const float* __restrict__ pos_sums, float* __restrict__ out) {
    __shared__ float s[RPE_L];              // 8 KB

    const int blocks_per_head = RPE_L / RPE_ROWS;   // 128
    const int h  = blockIdx.x / blocks_per_head;
    const int r0 = (blockIdx.x % blocks_per_head) * RPE_ROWS;
    const int t  = threadIdx.x;

    const float* src = pos_sums + (size_t)h * RPE_P;
    #pragma unroll
    for (int i = 0; i < RPE_L; i += 256) {
        int idx = i + t;
        s[idx] = (idx < RPE_P) ? src[idx] : 0.0f;
    }
    __syncthreads();

    float* orow = out + ((size_t)h * RPE_L + r0) * RPE_L;

    #pragma unroll
    for (int r = 0; r < RPE_ROWS; ++r) {
        const int i   = r0 + r;
        const int off = 1023 - i;           // rel index = clamp(j + off, 0, 2046)
        #pragma unroll
        for (int half = 0; half < 2; ++half) {
            const int j  = t * 4 + half * 1024;
            const int i0 = j + off;
            v4f v;
            v.x = s[min(max(i0 + 0, 0), RPE_P - 1)];
            v.y = s[min(max(i0 + 1, 0), RPE_P - 1)];
            v.z = s[min(max(i0 + 2, 0), RPE_P - 1)];
            v.w = s[min(max(i0 + 3, 0), RPE_P - 1)];
            // Streaming output (256 MB, never re-read): non-temporal b128 store.
            __builtin_nontemporal_store(v, (v4f*)(orow + (size_t)r * RPE_L + j));
        }
    }
}

extern "C" void kernel_launch(void* const* d_in, const int* in_sizes, int n_in,
                              void* d_out, int out_size, void* d_ws, size_t ws_size,
                              hipStream_t stream) {
    const float* E   = (const float*)d_in[0];   // (16, 2047, 64, 1) f32
    float*       out = (float*)d_out;            // (16, 2048, 2048) f32
    float*       ps  = (float*)d_ws;             // scratch: 16*2047 floats = 131 KB

    // Phase 1: 16*128 tiles, one wave each.
    rpe_reduce_wmma<<<RPE_H * 128, 32, 0, stream>>>(E, ps);

    // Phase 2: 16 heads * 128 row-groups.
    rpe_gather_rows<<<RPE_H * (RPE_L / RPE_ROWS), 256, 0, stream>>>(ps, out);
}